// RevEcho_37623913513160
// MI455X (gfx1250) — compile-verified
//
#include <hip/hip_runtime.h>
#include <cstddef>

// ---------------------------------------------------------------------------
// Compile-time replication of Python random.Random(42) echo schedule.
// ---------------------------------------------------------------------------
struct MT19937 {
    unsigned mt[624];
    int mti;
    constexpr MT19937() : mt{}, mti(625) {}
    constexpr void init_genrand(unsigned s) {
        mt[0] = s;
        for (mti = 1; mti < 624; mti++)
            mt[mti] = 1812433253u * (mt[mti - 1] ^ (mt[mti - 1] >> 30)) + (unsigned)mti;
    }
    constexpr void init_by_array(const unsigned* key, int klen) {
        init_genrand(19650218u);
        int i = 1, j = 0;
        int k = (624 > klen) ? 624 : klen;
        for (; k; k--) {
            mt[i] = (mt[i] ^ ((mt[i - 1] ^ (mt[i - 1] >> 30)) * 1664525u)) + key[j] + (unsigned)j;
            i++; j++;
            if (i >= 624) { mt[0] = mt[623]; i = 1; }
            if (j >= klen) j = 0;
        }
        for (k = 623; k; k--) {
            mt[i] = (mt[i] ^ ((mt[i - 1] ^ (mt[i - 1] >> 30)) * 1566083941u)) - (unsigned)i;
            i++;
            if (i >= 624) { mt[0] = mt[623]; i = 1; }
        }
        mt[0] = 0x80000000u;
        mti = 624;
    }
    constexpr unsigned next() {
        if (mti >= 624) {
            for (int kk = 0; kk < 624; kk++) {
                unsigned y = (mt[kk] & 0x80000000u) | (mt[(kk + 1) % 624] & 0x7fffffffu);
                unsigned v = mt[(kk + 397) % 624] ^ (y >> 1);
                if (y & 1u) v ^= 0x9908b0dfu;
                mt[kk] = v;
            }
            mti = 0;
        }
        unsigned y = mt[mti++];
        y ^= y >> 11;
        y ^= (y << 7) & 0x9d2c5680u;
        y ^= (y << 15) & 0xefc60000u;
        y ^= y >> 18;
        return y;
    }
    // CPython random_random(): 53-bit double in [0,1)
    constexpr double random() {
        unsigned a = next() >> 5, b = next() >> 6;
        return (a * 67108864.0 + b) * (1.0 / 9007199254740992.0);
    }
};

// constexpr exp via Cody-Waite range reduction + Taylor (double-accurate).
constexpr double c_exp(double x) {
    const double LN2_HI = 6.93147180369123816490e-01;
    const double LN2_LO = 1.90821492927058770002e-10;
    const double INV_LN2 = 1.44269504088896338700e+00;
    double t = x * INV_LN2;
    int n = (int)(t >= 0 ? t + 0.5 : t - 0.5);
    double r = (x - n * LN2_HI) - n * LN2_LO;
    double s = 0.0;
    for (int i = 20; i >= 1; i--) s = (s + 1.0) * (r / (double)i);
    double e = 1.0 + s;
    while (n > 0) { e *= 2.0; n--; }
    while (n < 0) { e *= 0.5; n++; }
    return e;
}
constexpr double pow10_c(double e) {
    return c_exp(e * 2.302585092994045684017991454684364207601);
}

constexpr int LEN  = 160000;
constexpr int NB   = 64;
constexpr int MAXT = 600;

struct TapTable {
    int   n;
    int   dmax;
    bool  overflow;
    int   d[MAXT];
    float a[MAXT];   // amplitudes with KEEP_CLEAN (0.1) folded in
};

constexpr TapTable make_taps() {
    TapTable T{};
    T.n = 0; T.dmax = 0; T.overflow = false;
    MT19937 rng;
    unsigned key[1] = {42u};
    rng.init_by_array(key, 1);
    double gate = rng.random(); (void)gate;              // PROBA gate (always passes)
    double initial     = rng.random() * 0.3;             // rng.random()*INITIAL
    double first_delay = 0.01 + (0.03 - 0.01) * rng.random();  // uniform(0.01,0.03)
    double rt60        = 0.3  + (1.3  - 0.3 ) * rng.random();  // uniform(0.3,1.3)
    for (int rep = 0; rep < 3; rep++) {
        double frac = 1.0, amp = initial;
        long long cum = 0;
        while (frac > 0.001) {
            double u  = -1.0 + 2.0 * rng.random();       // uniform(-1,1)
            double jj = 1.0 + 0.1 * u;                   // 1 + JITTER*u
            double prod = (jj * first_delay) * 16000.0;  // j*first_delay*SR (left assoc)
            long long delay = 1 + (long long)prod;       // 1 + int(...)
            if (delay > LEN) delay = LEN;
            cum += delay;
            if (cum > LEN) cum = LEN;
            if (cum < LEN) {
                if (T.n < MAXT) {
                    T.d[T.n] = (int)cum;
                    T.a[T.n] = (float)(0.1 * amp);       // fold KEEP_CLEAN
                    if ((int)cum > T.dmax) T.dmax = (int)cum;
                    T.n++;
                } else {
                    T.overflow = true;
                }
            }
            double u2  = -1.0 + 2.0 * rng.random();
            double j2  = 1.0 + 0.1 * u2;
            double att = pow10_c((((-3.0) * j2) * first_delay) / rt60);
            amp  *= att;
            frac *= att;
        }
    }
    return T;
}

constexpr TapTable HT = make_taps();
static_assert(!HT.overflow, "tap table overflow");
static_assert(HT.n > 0, "no taps");

constexpr int NTAPS  = HT.n;
constexpr int BLOCK  = 256;
constexpr int ROWS   = 32;
constexpr int TILE   = BLOCK * ROWS;                 // 8192 outputs / workgroup
constexpr int TILES  = (LEN + TILE - 1) / TILE;      // 20
constexpr int DMAX4  = (HT.dmax + 3) & ~3;           // 16B-aligned window offset
constexpr int REGION = DMAX4 + TILE;                 // staged floats per tile
static_assert(REGION * 4 <= 160 * 1024, "LDS window too large");
static_assert((LEN & 3) == 0, "alignment reasoning needs LEN % 4 == 0");

// Packed tap: one s_load_b64 per tap in the inner loop.
struct Tap { int d; float a; };
struct TapPacked { Tap t[MAXT]; };
constexpr TapPacked pack_taps() {
    TapPacked P{};
    for (int i = 0; i < MAXT; i++) {
        P.t[i].d = (i < HT.n) ? HT.d[i] : 0;
        P.t[i].a = (i < HT.n) ? HT.a[i] : 0.0f;
    }
    return P;
}
__constant__ TapPacked c_taps = pack_taps();

// ---------------------------------------------------------------------------
// CDNA5 async global->LDS staging (ASYNCcnt path), with portable fallback.
// Probe-confirmed signature: (v4i AS1* src, v4i AS3* dst, Ii offset, Ii cpol).
// ---------------------------------------------------------------------------
#if defined(__AMDGCN__) && __has_builtin(__builtin_amdgcn_global_load_async_to_lds_b128)
#define HAVE_ASYNC 1
#else
#define HAVE_ASYNC 0
#endif

#if HAVE_ASYNC
typedef int ai4_t __attribute__((ext_vector_type(4)));
typedef __attribute__((address_space(1))) ai4_t* gv4_p;
typedef __attribute__((address_space(3))) ai4_t* lv4_p;
#endif

__device__ __forceinline__ void stage_b128(float* l, const float* g) {
#if HAVE_ASYNC
    float* gnc = const_cast<float*>(g);
    __builtin_amdgcn_global_load_async_to_lds_b128((gv4_p)gnc, (lv4_p)l, 0, 0);
#else
    float4 v = *(const float4*)g;
    *(float4*)l = v;
#endif
}

__device__ __forceinline__ void wait_async0() {
#if HAVE_ASYNC
#if __has_builtin(__builtin_amdgcn_s_wait_asynccnt)
    __builtin_amdgcn_s_wait_asynccnt(0);
#else
    asm volatile("s_wait_asynccnt 0" ::: "memory");
#endif
#endif
}

__device__ __forceinline__ int imax(int a, int b) { return a > b ? a : b; }
__device__ __forceinline__ int imin(int a, int b) { return a < b ? a : b; }

// ---------------------------------------------------------------------------
// Kernel: one workgroup = one (batch, TILE) slab. Stage [t0-DMAX4, t0+TILE)
// into LDS once via async b128 copies (workgroup-uniform phase bounds, so
// edge handling is scalar-branched), then each thread accumulates ROWS
// outputs strided by BLOCK: per-tap LDS reads are lane-consecutive
// (bank-conflict-free) and one s_load_b64 tap fetch feeds 32 v_fmac.
// ---------------------------------------------------------------------------
__global__ __launch_bounds__(BLOCK)
void revecho_kernel(const float* __restrict__ x, float* __restrict__ y) {
    __shared__ float lds[REGION];

    const int b     = blockIdx.y;
    const int tid   = (int)threadIdx.x;
    const int tile0 = blockIdx.x * TILE;
    const int gbase = tile0 - DMAX4;                 // multiple of 4
    const float* __restrict__ row = x + (size_t)b * LEN;

    // ---- uniform staging bounds (all multiples of 4 elements) ----
    const int lo = imin(REGION, imax(0, -gbase));        // zeros before signal
    const int hi = imax(lo, imin(REGION, LEN - gbase));  // zeros after signal

    if (lo > 0) {                                    // uniform branch (first tiles)
        for (int i = tid; i < lo; i += BLOCK) lds[i] = 0.0f;
    }
    if (hi < REGION) {                               // uniform branch (last tile)
        for (int i = hi + tid; i < REGION; i += BLOCK) lds[i] = 0.0f;
    }
    for (int i = lo + tid * 4; i < hi; i += BLOCK * 4)
        stage_b128(&lds[i], row + (gbase + i));

    wait_async0();
    __syncthreads();

    // ---- sparse FIR from LDS ----
    float acc[ROWS];
#pragma unroll
    for (int r = 0; r < ROWS; r++)
        acc[r] = lds[DMAX4 + r * BLOCK + tid];       // clean signal term

#pragma unroll 2
    for (int k = 0; k < NTAPS; k++) {
        const Tap tp = c_taps.t[k];                  // one s_load_b64
        const int base = DMAX4 + tid - tp.d;
#pragma unroll
        for (int r = 0; r < ROWS; r++)
            acc[r] += tp.a * lds[base + r * BLOCK];
    }

    // ---- store (uniform fast path for full tiles) ----
    float* __restrict__ out = y + (size_t)b * LEN + tile0;
    if (tile0 + TILE <= LEN) {
#pragma unroll
        for (int r = 0; r < ROWS; r++)
            out[r * BLOCK + tid] = acc[r];
    } else {
#pragma unroll
        for (int r = 0; r < ROWS; r++) {
            const int t = r * BLOCK + tid;
            if (tile0 + t < LEN) out[t] = acc[r];
        }
    }
}

// ---------------------------------------------------------------------------
extern "C" void kernel_launch(void* const* d_in, const int* in_sizes, int n_in,
                              void* d_out, int out_size, void* d_ws, size_t ws_size,
                              hipStream_t stream) {
    (void)in_sizes; (void)n_in; (void)out_size; (void)d_ws; (void)ws_size;
    const float* x = (const float*)d_in[0];
    float*       y = (float*)d_out;
    dim3 grid(TILES, NB);
    revecho_kernel<<<grid, dim3(BLOCK), 0, stream>>>(x, y);
}